// get_local_area_66743791780161
// MI455X (gfx1250) — compile-verified
//
#include <hip/hip_runtime.h>
#include <hip/hip_bf16.h>
#include <stddef.h>

#define BATCH   8
#define NPTS    8192
#define NPOINT  1024
#define NSAMPLE 32
#define DIMF    192          /* 64*3 flattened feature channels */
#define RADIUS2 0.04f

typedef __attribute__((ext_vector_type(2))) float v2f;
typedef __attribute__((ext_vector_type(8))) float v8f;

// ---------------------------------------------------------------------------
// Stage 1: furthest point sampling. One 1024-thread block per batch; each
// thread keeps its 8 points (t, t+1024, ...) and running min-dist in regs.
// LDS tree argmax with first-index tie-break (matches jnp.argmax).
// ---------------------------------------------------------------------------
__global__ __launch_bounds__(1024)
void fps_kernel(const float* __restrict__ xyz, int* __restrict__ fps_ids,
                float* __restrict__ new_xyz)
{
  const int b = blockIdx.x;
  const int t = threadIdx.x;
  const float* bx = xyz + (size_t)b * NPTS * 3;

  float px[8], py[8], pz[8], md[8];
#pragma unroll
  for (int k = 0; k < 8; ++k) {
    const int p = t + (k << 10);
    px[k] = bx[p * 3 + 0];
    py[k] = bx[p * 3 + 1];
    pz[k] = bx[p * 3 + 2];
    md[k] = 1.0e10f;
  }

  __shared__ float s_val[1024];
  __shared__ int   s_idx[1024];
  __shared__ float s_last[3];

  if (t == 0) {
    fps_ids[b * NPOINT] = 0;
    new_xyz[((size_t)b * NPOINT) * 3 + 0] = bx[0];
    new_xyz[((size_t)b * NPOINT) * 3 + 1] = bx[1];
    new_xyz[((size_t)b * NPOINT) * 3 + 2] = bx[2];
    s_last[0] = bx[0]; s_last[1] = bx[1]; s_last[2] = bx[2];
  }
  __syncthreads();

  for (int j = 1; j < NPOINT; ++j) {
    const float lx = s_last[0], ly = s_last[1], lz = s_last[2];
    float bv = -1.0f;
    int   bi = 0;
#pragma unroll
    for (int k = 0; k < 8; ++k) {
      const float dx = px[k] - lx, dy = py[k] - ly, dz = pz[k] - lz;
      const float d  = dx * dx + dy * dy + dz * dz;
      md[k] = fminf(md[k], d);
      const int p = t + (k << 10);
      if (md[k] > bv || (md[k] == bv && p < bi)) { bv = md[k]; bi = p; }
    }
    s_val[t] = bv;
    s_idx[t] = bi;
    __syncthreads();
    for (int off = 512; off > 0; off >>= 1) {
      if (t < off) {
        const float ov = s_val[t + off];
        const int   oi = s_idx[t + off];
        if (ov > s_val[t] || (ov == s_val[t] && oi < s_idx[t])) {
          s_val[t] = ov; s_idx[t] = oi;
        }
      }
      __syncthreads();
    }
    const int nxt = s_idx[0];
    if (t == (nxt & 1023)) {            // owner thread broadcasts coords
      const int k = nxt >> 10;
      s_last[0] = px[k]; s_last[1] = py[k]; s_last[2] = pz[k];
      new_xyz[((size_t)b * NPOINT + j) * 3 + 0] = px[k];
      new_xyz[((size_t)b * NPOINT + j) * 3 + 1] = py[k];
      new_xyz[((size_t)b * NPOINT + j) * 3 + 2] = pz[k];
      fps_ids[b * NPOINT + j] = nxt;
    }
    __syncthreads();
  }
}

// ---------------------------------------------------------------------------
// Stage 2: gather new_fts[b][c][j] = fts[b][c][fps_ids[b][j]]
// ---------------------------------------------------------------------------
__global__ __launch_bounds__(256)
void gather_fts_kernel(const float* __restrict__ fts, const int* __restrict__ fps_ids,
                       float* __restrict__ new_fts)
{
  const int i = blockIdx.x * 256 + threadIdx.x;
  const int total = BATCH * DIMF * NPOINT;
  if (i >= total) return;
  const int j = i & (NPOINT - 1);
  const int c = (i >> 10) % DIMF;
  const int b = i / (NPOINT * DIMF);
  const int id = fps_ids[b * NPOINT + j];
  new_fts[i] = fts[((size_t)b * DIMF + c) * NPTS + id];
}

// ---------------------------------------------------------------------------
// Stage 3: ball query via WMMA. One wave = 16 centers.
// d2 tile (16 centers x 16 points) = V_WMMA_F32_16X16X4_F32 with
//   A row m = (-2cx, -2cy, -2cz, |c|^2)   (K padded to 4)
//   B col n = ( px,   py,   pz,   1   )
//   C[.][n] = |p_n|^2
// Accumulator VGPR g holds rows g (lanes 0-15) and g+8 (lanes 16-31), so
// ballot_w32(D[g] < r^2) yields both rows' 16-bit hit masks as a uniform
// scalar. Lane m selects its row mask via a cndmask tree and extracts hit
// indices with ctz (iterating only over actual hits), preserving ascending
// index order to match the reference's top-k over where(d2<r^2, idx, n).
// Fill = first hit (or 0 if none). Ballot early-exit when all rows full.
// ---------------------------------------------------------------------------
__global__ __launch_bounds__(256)
void ball_query_kernel(const float* __restrict__ xyz,
                       const float* __restrict__ new_xyz,
                       int* __restrict__ idx_out)
{
  const int lane  = threadIdx.x & 31;
  const int wslot = threadIdx.x >> 5;
  const int wid   = blockIdx.x * 8 + wslot;      // 0..511
  const int b     = wid >> 6;                    // wid / (NPOINT/16)
  const int mt    = wid & 63;                    // center tile within batch
  const int m     = lane & 15;

  // A matrix: 16x4 f32; lanes 0-15 hold (K0,K1), lanes 16-31 hold (K2,K3)
  const float* cen = new_xyz + ((size_t)b * NPOINT + mt * 16 + m) * 3;
  const float cx = cen[0], cy = cen[1], cz = cen[2];
  const float cn2 = cx * cx + cy * cy + cz * cz;
  v2f A;
  if (lane < 16) { A.x = -2.0f * cx; A.y = -2.0f * cy; }
  else           { A.x = -2.0f * cz; A.y = cn2;        }

  const float* bx = xyz + (size_t)b * NPTS * 3;
  int* row = idx_out + ((size_t)b * NPOINT + mt * 16 + m) * NSAMPLE;

  int cnt = 0, first = 0;
  for (int nt = 0; nt < NPTS / 16; ++nt) {
    const int n0 = nt * 16;
    // B matrix 4x16: lanes 0-15 hold (K0,K1)=(px,py), lanes 16-31 (K2,K3)=(pz,1)
    const float* p3 = bx + (size_t)(n0 + m) * 3;
    const float qx = p3[0], qy = p3[1], qz = p3[2];
    const float pn2 = qx * qx + qy * qy + qz * qz;
    v2f Bv;
    if (lane < 16) { Bv.x = qx; Bv.y = qy;   }
    else           { Bv.x = qz; Bv.y = 1.0f; }
    v8f C;
#pragma unroll
    for (int g = 0; g < 8; ++g) C[g] = pn2;   // |p_n|^2, same for every row

    const v8f D = __builtin_amdgcn_wmma_f32_16x16x4_f32(
        /*neg_a=*/false, A, /*neg_b=*/false, Bv,
        /*c_mod=*/(short)0, C, /*reuse_a=*/false, /*reuse_b=*/false);

    // Whole-tile radius test -> 8 uniform 32-bit masks (rows g | g+8).
    const unsigned bl0 = __builtin_amdgcn_ballot_w32(D[0] < RADIUS2);
    const unsigned bl1 = __builtin_amdgcn_ballot_w32(D[1] < RADIUS2);
    const unsigned bl2 = __builtin_amdgcn_ballot_w32(D[2] < RADIUS2);
    const unsigned bl3 = __builtin_amdgcn_ballot_w32(D[3] < RADIUS2);
    const unsigned bl4 = __builtin_amdgcn_ballot_w32(D[4] < RADIUS2);
    const unsigned bl5 = __builtin_amdgcn_ballot_w32(D[5] < RADIUS2);
    const unsigned bl6 = __builtin_amdgcn_ballot_w32(D[6] < RADIUS2);
    const unsigned bl7 = __builtin_amdgcn_ballot_w32(D[7] < RADIUS2);

    if (lane < 16 && cnt < NSAMPLE) {
      // Select row m's ballot word (uniform values, divergent selector).
      const int g = m & 7;
      const unsigned t0 = (g & 1) ? bl1 : bl0;
      const unsigned t1 = (g & 1) ? bl3 : bl2;
      const unsigned t2 = (g & 1) ? bl5 : bl4;
      const unsigned t3 = (g & 1) ? bl7 : bl6;
      const unsigned u0 = (g & 2) ? t1 : t0;
      const unsigned u1 = (g & 2) ? t3 : t2;
      const unsigned sw = (g & 4) ? u1 : u0;
      unsigned hm = (m < 8) ? (sw & 0xffffu) : (sw >> 16);
      while (hm != 0u && cnt < NSAMPLE) {
        const int q = __builtin_ctz(hm);
        hm &= hm - 1u;
        const int pid = n0 + q;
        if (cnt == 0) first = pid;
        row[cnt] = pid;
        ++cnt;
      }
    }
    const unsigned done =
        __builtin_amdgcn_ballot_w32((lane >= 16) || (cnt >= NSAMPLE));
    if (done == 0xFFFFFFFFu) break;
  }
  if (lane < 16) {
    const int fill = (cnt == 0) ? 0 : first;
    for (int s = cnt; s < NSAMPLE; ++s) row[s] = fill;
  }
}

// ---------------------------------------------------------------------------
// Stage 4a: group_xyz[b][j][s][:] = xyz[b][idx][:] - new_xyz[b][j][:]
// ---------------------------------------------------------------------------
__global__ __launch_bounds__(256)
void group_xyz_kernel(const float* __restrict__ xyz, const float* __restrict__ new_xyz,
                      const int* __restrict__ idx, float* __restrict__ out)
{
  const int i = blockIdx.x * 256 + threadIdx.x;   // (b, j, s) flat, 262144 total
  const int b = i >> 15;
  const int j = (i >> 5) & (NPOINT - 1);
  const int id = idx[i];
  const float* p = xyz + ((size_t)b * NPTS + id) * 3;
  const float* c = new_xyz + ((size_t)b * NPOINT + j) * 3;
  float* o = out + (size_t)i * 3;
  o[0] = p[0] - c[0];
  o[1] = p[1] - c[1];
  o[2] = p[2] - c[2];
}

// ---------------------------------------------------------------------------
// Stage 4b: group_fts[b][c][j][s] = fts[b][c][idx[b][j][s]], float4 over s.
// ---------------------------------------------------------------------------
__global__ __launch_bounds__(256)
void group_fts_kernel(const float* __restrict__ fts, const int* __restrict__ idx,
                      float* __restrict__ out)
{
  const long long i = ((long long)blockIdx.x * 256 + threadIdx.x) * 4;
  const int s = (int)(i & (NSAMPLE - 1));
  const int j = (int)((i >> 5) & (NPOINT - 1));
  const int c = (int)((i >> 15) % DIMF);
  const int b = (int)(i / ((long long)NSAMPLE * NPOINT * DIMF));
  const int4 id4 = *(const int4*)(idx + ((size_t)b * NPOINT + j) * NSAMPLE + s);
  const float* src = fts + ((size_t)b * DIMF + c) * NPTS;
  float4 v;
  v.x = src[id4.x];
  v.y = src[id4.y];
  v.z = src[id4.z];
  v.w = src[id4.w];
  *(float4*)(out + i) = v;
}

// ---------------------------------------------------------------------------
extern "C" void kernel_launch(void* const* d_in, const int* in_sizes, int n_in,
                              void* d_out, int out_size, void* d_ws, size_t ws_size,
                              hipStream_t stream)
{
  const float* xyz = (const float*)d_in[0];   // (B, N, 3)
  const float* fts = (const float*)d_in[1];   // (B, 192, N)

  float* out  = (float*)d_out;
  float* gxyz = out;                                              // 786432
  float* gfts = gxyz + (size_t)BATCH * NPOINT * NSAMPLE * 3;      // 50331648
  float* nxyz = gfts + (size_t)BATCH * DIMF * NPOINT * NSAMPLE;   // 24576
  float* nfts = nxyz + (size_t)BATCH * NPOINT * 3;                // 1572864

  int* fps_ids = (int*)d_ws;                           // B*NPOINT ints
  int* bidx    = fps_ids + BATCH * NPOINT;             // B*NPOINT*NSAMPLE ints

  fps_kernel<<<BATCH, 1024, 0, stream>>>(xyz, fps_ids, nxyz);
  gather_fts_kernel<<<(BATCH * DIMF * NPOINT) / 256, 256, 0, stream>>>(fts, fps_ids, nfts);
  ball_query_kernel<<<(BATCH * (NPOINT / 16)) / 8, 256, 0, stream>>>(xyz, nxyz, bidx);
  group_xyz_kernel<<<(BATCH * NPOINT * NSAMPLE) / 256, 256, 0, stream>>>(xyz, nxyz, bidx, gxyz);
  group_fts_kernel<<<((size_t)BATCH * DIMF * NPOINT * NSAMPLE / 4) / 256, 256, 0, stream>>>(fts, bidx, gfts);
}